// GAT_14740327760516
// MI455X (gfx1250) — compile-verified
//
#include <hip/hip_runtime.h>
#include <hip/hip_bf16.h>

#define NNODES 50000
#define NEDGES 400000
#define HD 256
#define NH 4
#define DH 64

typedef float v2f __attribute__((ext_vector_type(2)));
typedef float v8f __attribute__((ext_vector_type(8)));

// ---------------- utility fills ----------------
__global__ __launch_bounds__(256) void fill_f32(float* __restrict__ p, float v, int n) {
  int i = blockIdx.x * 256 + threadIdx.x;
  if (i < n) p[i] = v;
}

__global__ __launch_bounds__(256) void fill_u32(unsigned* __restrict__ p, unsigned v, int n) {
  int i = blockIdx.x * 256 + threadIdx.x;
  if (i < n) p[i] = v;
}

// ---------------- x = emb[nids] ----------------
__global__ __launch_bounds__(256) void gather_rows(const float* __restrict__ emb,
                                                   const int* __restrict__ nids,
                                                   float* __restrict__ x, int total) {
  int i = blockIdx.x * 256 + threadIdx.x;
  if (i >= total) return;
  int n = i >> 8;
  int c = i & (HD - 1);
  x[i] = emb[(size_t)nids[n] * HD + c];
}

// ---------------- f = x @ W via V_WMMA_F32_16X16X4_F32 ----------------
// One wave computes one 16x16 tile of F. 8 waves per 256-thread block.
// A (16x4 f32): lane<16 -> row M=lane, K=k..k+1 ; lane>=16 -> row M=lane-16, K=k+2..k+3
// B (4x16 f32): lane<16 -> col N=lane, rows K=k,k+1 ; lane>=16 -> col N=lane-16, rows K=k+2,k+3
// C/D (16x16 f32): VGPR r holds row m0 + r + 8*(lane>=16), col n0 + (lane&15)
__global__ __launch_bounds__(256) void gemm_wmma_f32(const float* __restrict__ X,
                                                     const float* __restrict__ W,
                                                     float* __restrict__ F, int nrows) {
  const int lane = threadIdx.x & 31;
  const int wave = threadIdx.x >> 5;
  const int tile = blockIdx.x * 8 + wave;
  const int nTiles = HD / 16;                 // 16 tiles across columns
  const int mTile = tile / nTiles;
  const int nTile = tile - mTile * nTiles;
  const int m0 = mTile * 16;
  const int n0 = nTile * 16;
  if (m0 >= nrows) return;                    // wave-uniform; EXEC stays all-ones

  const int half = lane >> 4;                 // 0 or 1
  const int l = lane & 15;

  const float* __restrict__ xrow = X + (size_t)(m0 + l) * HD + 2 * half;
  const float* __restrict__ wcol = W + (size_t)(2 * half) * HD + n0 + l;

  v8f c = {};
#pragma unroll 4
  for (int k = 0; k < HD; k += 4) {
    v2f a, b;
    a[0] = xrow[k];                           // contiguous K pair -> b64 load
    a[1] = xrow[k + 1];
    b[0] = wcol[k * HD];                      // row k (+2*half)
    b[1] = wcol[k * HD + HD];                 // row k+1 (+2*half)
    c = __builtin_amdgcn_wmma_f32_16x16x4_f32(false, a, false, b, (short)0, c,
                                              false, false);
  }

  float* __restrict__ frow = F + (size_t)(m0 + 8 * half) * HD + n0 + l;
#pragma unroll
  for (int r = 0; r < 8; ++r) frow[(size_t)r * HD] = c[r];
}

// ---------------- el/er = einsum('nhd,hd->nh') ----------------
__global__ __launch_bounds__(256) void attn_logits(const float* __restrict__ F,
                                                   const float* __restrict__ al,
                                                   const float* __restrict__ ar,
                                                   float* __restrict__ el,
                                                   float* __restrict__ er, int total) {
  int i = blockIdx.x * 256 + threadIdx.x;     // over N*NH
  if (i >= total) return;
  int n = i >> 2;
  int h = i & (NH - 1);
  const float4* f = (const float4*)(F + (size_t)n * HD + h * DH);
  const float4* a = (const float4*)(al + h * DH);
  const float4* b = (const float4*)(ar + h * DH);
  float sl = 0.f, sr = 0.f;
#pragma unroll
  for (int j = 0; j < DH / 4; ++j) {
    float4 fv = f[j], av = a[j], bv = b[j];
    sl += fv.x * av.x + fv.y * av.y + fv.z * av.z + fv.w * av.w;
    sr += fv.x * bv.x + fv.y * bv.y + fv.z * bv.z + fv.w * bv.w;
  }
  el[i] = sl;
  er[i] = sr;
}

// order-preserving f32 -> u32 encoding for atomic max
__device__ __forceinline__ unsigned enc_f32(float v) {
  unsigned b = __float_as_uint(v);
  return (b & 0x80000000u) ? ~b : (b | 0x80000000u);
}
__device__ __forceinline__ float dec_f32(unsigned e) {
  unsigned b = (e & 0x80000000u) ? (e & 0x7FFFFFFFu) : ~e;
  return __uint_as_float(b);
}

// ---------------- edge pass 1: leaky-relu logit + segment max ----------------
__global__ __launch_bounds__(256) void edge_logit_max(const float* __restrict__ el,
                                                      const float* __restrict__ er,
                                                      const int* __restrict__ src,
                                                      const int* __restrict__ dst,
                                                      float* __restrict__ elog,
                                                      unsigned* __restrict__ mEnc,
                                                      int total) {
  int i = blockIdx.x * 256 + threadIdx.x;     // over E*NH
  if (i >= total) return;
  int e = i >> 2;
  int h = i & (NH - 1);
  int sN = src[e], dN = dst[e];
  float v = el[sN * NH + h] + er[dN * NH + h];
  v = (v > 0.f) ? v : 0.2f * v;               // leaky relu, slope 0.2
  elog[i] = v;
  atomicMax(mEnc + dN * NH + h, enc_f32(v));
}

// ---------------- edge pass 2: ex = exp(e - m[dst]); segment sum ----------------
__global__ __launch_bounds__(256) void edge_exp_sum(float* __restrict__ elog,
                                                    const int* __restrict__ dst,
                                                    const unsigned* __restrict__ mEnc,
                                                    float* __restrict__ ssum, int total) {
  int i = blockIdx.x * 256 + threadIdx.x;     // over E*NH
  if (i >= total) return;
  int e = i >> 2;
  int h = i & (NH - 1);
  int dN = dst[e];
  float m = dec_f32(mEnc[dN * NH + h]);
  float x = __expf(elog[i] - m);
  elog[i] = x;                                // overwrite logits with exp values
  atomicAdd(ssum + dN * NH + h, x);
}

// ---------------- edge aggregate: acc[dst] += ex * f[src] ----------------
// one wave per edge; lane owns 8 contiguous channels (head = lane/8)
__global__ __launch_bounds__(256) void edge_aggregate(const float* __restrict__ F,
                                                      const float* __restrict__ ex,
                                                      const int* __restrict__ src,
                                                      const int* __restrict__ dst,
                                                      float* __restrict__ acc, int E) {
  int e = blockIdx.x * 8 + (threadIdx.x >> 5);
  if (e >= E) return;
  int lane = threadIdx.x & 31;
  int sN = src[e], dN = dst[e];
  float w = ex[e * NH + (lane >> 3)];
  const float* fs = F + (size_t)sN * HD + lane * 8;
  float* ad = acc + (size_t)dN * HD + lane * 8;
  float4 f0 = *(const float4*)(fs);
  float4 f1 = *(const float4*)(fs + 4);
  atomicAdd(ad + 0, w * f0.x);
  atomicAdd(ad + 1, w * f0.y);
  atomicAdd(ad + 2, w * f0.z);
  atomicAdd(ad + 3, w * f0.w);
  atomicAdd(ad + 4, w * f1.x);
  atomicAdd(ad + 5, w * f1.y);
  atomicAdd(ad + 6, w * f1.z);
  atomicAdd(ad + 7, w * f1.w);
}

// ---------------- finalize: out = acc/s + b (optional ELU) ----------------
__global__ __launch_bounds__(256) void finalize(const float* __restrict__ acc,
                                                const float* __restrict__ ssum,
                                                const float* __restrict__ bias,
                                                float* __restrict__ out, int total,
                                                int applyElu) {
  int i = blockIdx.x * 256 + threadIdx.x;     // over N*HD
  if (i >= total) return;
  int n = i >> 8;
  int c = i & (HD - 1);
  int h = c >> 6;
  float sv = ssum[n * NH + h];
  float v = (sv > 0.f) ? (acc[i] / sv) : 0.f; // empty dst -> 0 (+bias)
  v += bias[c];
  if (applyElu) v = (v > 0.f) ? v : (__expf(v) - 1.0f);
  out[i] = v;
}

extern "C" void kernel_launch(void* const* d_in, const int* in_sizes, int n_in,
                              void* d_out, int out_size, void* d_ws, size_t ws_size,
                              hipStream_t stream) {
  (void)in_sizes; (void)n_in; (void)out_size; (void)ws_size;

  const int* nids = (const int*)d_in[0];
  const int* esrc = (const int*)d_in[1];
  const int* edst = (const int*)d_in[2];
  const float* emb = (const float*)d_in[3];
  const float* Ws[2]  = {(const float*)d_in[4], (const float*)d_in[8]};
  const float* als[2] = {(const float*)d_in[5], (const float*)d_in[9]};
  const float* ars[2] = {(const float*)d_in[6], (const float*)d_in[10]};
  const float* bs[2]  = {(const float*)d_in[7], (const float*)d_in[11]};

  float* ws = (float*)d_ws;
  size_t off = 0;
  float* x = ws + off;      off += (size_t)NNODES * HD;   // layer input
  float* f = ws + off;      off += (size_t)NNODES * HD;   // fc output
  float* acc = ws + off;    off += (size_t)NNODES * HD;   // aggregation accumulator
  float* el = ws + off;     off += (size_t)NNODES * NH;
  float* er = ws + off;     off += (size_t)NNODES * NH;
  float* ssum = ws + off;   off += (size_t)NNODES * NH;
  unsigned* mEnc = (unsigned*)(ws + off); off += (size_t)NNODES * NH;
  float* eE = ws + off;     off += (size_t)NEDGES * NH;   // logits, then exp values

  const int NC = NNODES * HD;                  // 12.8M
  const int gNC   = (NC + 255) / 256;
  const int gNHn  = (NNODES * NH + 255) / 256;
  const int gEH   = (NEDGES * NH + 255) / 256;
  const int gE8   = (NEDGES + 7) / 8;
  const int gGemm = ((NNODES / 16) * (HD / 16) + 7) / 8;  // 6250 blocks, exact

  gather_rows<<<gNC, 256, 0, stream>>>(emb, nids, x, NC);

  for (int L = 0; L < 2; ++L) {
    fill_f32<<<gNC, 256, 0, stream>>>(acc, 0.f, NC);
    fill_f32<<<gNHn, 256, 0, stream>>>(ssum, 0.f, NNODES * NH);
    fill_u32<<<gNHn, 256, 0, stream>>>(mEnc, 0x007FFFFFu /* enc(-inf) */, NNODES * NH);

    gemm_wmma_f32<<<gGemm, 256, 0, stream>>>(x, Ws[L], f, NNODES);
    attn_logits<<<gNHn, 256, 0, stream>>>(f, als[L], ars[L], el, er, NNODES * NH);
    edge_logit_max<<<gEH, 256, 0, stream>>>(el, er, esrc, edst, eE, mEnc, NEDGES * NH);
    edge_exp_sum<<<gEH, 256, 0, stream>>>(eE, edst, mEnc, ssum, NEDGES * NH);
    edge_aggregate<<<gE8, 256, 0, stream>>>(f, eE, esrc, edst, acc, NEDGES);
    finalize<<<gNC, 256, 0, stream>>>(acc, ssum, bs[L],
                                      (L == 0) ? x : (float*)d_out, NC, (L == 0) ? 1 : 0);
  }
}